// ChannelGate_24661702214096
// MI455X (gfx1250) — compile-verified
//
#include <hip/hip_runtime.h>
#include <hip/hip_bf16.h>
#include <math.h>

// ---- problem constants (from the reference) ----
#define BB   32
#define CC   512
#define HH   56
#define WW   56
#define HWP  (HH*WW)          // 3136
#define HW4  (HWP/4)          // 784
#define RED  32               // C/16
#define BN_EPS 1e-5f

typedef float v2f __attribute__((ext_vector_type(2)));
typedef float v8f __attribute__((ext_vector_type(8)));

__device__ __forceinline__ float sigmoidf_(float t) {
    return 1.0f / (1.0f + __expf(-t));
}

// Correct float atomic-max for mixed-sign values (dest initialized to -inf).
__device__ __forceinline__ void atomicMaxF(float* addr, float val) {
    if (val >= 0.0f) atomicMax((int*)addr, __float_as_int(val));
    else             atomicMin((unsigned int*)addr, __float_as_uint(val));
}

// ---------------------------------------------------------------------------
// K0: init channel accumulators
// ---------------------------------------------------------------------------
__global__ __launch_bounds__(256) void k_init(float* __restrict__ ch_sum,
                                              float* __restrict__ ch_max) {
    int i = blockIdx.x * 256 + threadIdx.x;
    if (i < BB * CC) {
        ch_sum[i] = 0.0f;
        ch_max[i] = -__builtin_inff();
    }
}

// ---------------------------------------------------------------------------
// K1: single pass over x producing
//   per-pixel  mean/max over C  (thread-owned, direct store)
//   per-channel sum/max over HW (wave shuffle reduce -> LDS -> global atomics)
// grid: (ceil(HW/256), B), block: 256 (8 wave32)
// ---------------------------------------------------------------------------
__global__ __launch_bounds__(256) void k_stats(const float* __restrict__ x,
                                               float* __restrict__ ch_sum,
                                               float* __restrict__ ch_max,
                                               float* __restrict__ ss1,
                                               float* __restrict__ ss2) {
    __shared__ float lsum[8 * CC];
    __shared__ float lmax[8 * CC];

    const int tid  = threadIdx.x;
    const int wv   = tid >> 5;
    const int lane = tid & 31;
    const int b    = blockIdx.y;
    const int p    = blockIdx.x * 256 + tid;
    const bool valid = (p < HWP);

    const float* __restrict__ xb = x + (size_t)b * CC * HWP;

    float psum = 0.0f;
    float pmax = -__builtin_inff();

    for (int c = 0; c < CC; ++c) {
        float v  = valid ? xb[(size_t)c * HWP + p] : 0.0f;
        float vm = valid ? v : -__builtin_inff();
        psum += v;
        pmax  = fmaxf(pmax, vm);
        // wave32 butterfly reduce (sum + max)
        float ws = v, wm = vm;
        #pragma unroll
        for (int off = 16; off > 0; off >>= 1) {
            ws += __shfl_xor(ws, off, 32);
            wm  = fmaxf(wm, __shfl_xor(wm, off, 32));
        }
        if (lane == 0) { lsum[wv * CC + c] = ws; lmax[wv * CC + c] = wm; }
    }

    if (valid) {
        ss1[(size_t)b * HWP + p] = psum * (1.0f / (float)CC);
        ss2[(size_t)b * HWP + p] = pmax;
    }

    __syncthreads();
    for (int c = tid; c < CC; c += 256) {
        float s = 0.0f, m = -__builtin_inff();
        #pragma unroll
        for (int w = 0; w < 8; ++w) {
            s += lsum[w * CC + c];
            m  = fmaxf(m, lmax[w * CC + c]);
        }
        atomicAdd(&ch_sum[b * CC + c], s);
        atomicMaxF(&ch_max[b * CC + c], m);
    }
}

// ---------------------------------------------------------------------------
// K2: per-batch: finalize sc1/sc2, sigmoid(sc1*sc2), bitonic-sort both arrays
//     descending, interleave top halves -> squeeze_all[b, :]
// grid: B, block: 256
// ---------------------------------------------------------------------------
__global__ __launch_bounds__(256) void k_sort(const float* __restrict__ ch_sum,
                                              const float* __restrict__ ch_max,
                                              float* __restrict__ sig_sc,
                                              float* __restrict__ sq_all) {
    __shared__ float a1[CC];
    __shared__ float a2[CC];
    const int b = blockIdx.x, tid = threadIdx.x;

    for (int c = tid; c < CC; c += 256) {
        float s1 = ch_sum[b * CC + c] * (1.0f / (float)HWP);  // mean over HW
        float s2 = ch_max[b * CC + c];
        sig_sc[b * CC + c] = sigmoidf_(s1 * s2);
        a1[c] = s1;
        a2[c] = s2;
    }
    __syncthreads();

    // bitonic sort, overall DESCENDING, both arrays in lockstep
    for (int k = 2; k <= CC; k <<= 1) {
        for (int j = k >> 1; j > 0; j >>= 1) {
            for (int i = tid; i < CC; i += 256) {
                int ixj = i ^ j;
                if (ixj > i) {
                    bool up = ((i & k) == 0);     // "up" blocks sort descending
                    float u = a1[i], v = a1[ixj];
                    if (up ? (u < v) : (u > v)) { a1[i] = v; a1[ixj] = u; }
                    float s = a2[i], t = a2[ixj];
                    if (up ? (s < t) : (s > t)) { a2[i] = t; a2[ixj] = s; }
                }
            }
            __syncthreads();
        }
    }

    // interleave: [t1[0], t2[0], t1[1], t2[1], ...]
    if (tid < CC / 2) {
        sq_all[b * CC + 2 * tid + 0] = a1[tid];
        sq_all[b * CC + 2 * tid + 1] = a2[tid];
    }
}

// ---------------------------------------------------------------------------
// K3: MLP gate via V_WMMA_F32_16X16X4_F32 (fp32 matrix path)
//   H[32x32]   = relu( S[32x512] @ W1^T[512x32] + b1 )
//   O[32x512]  =       H[32x32]  @ W2^T[32x512] + b2
//   sq_w[b,c]  = relu( O[b,c] + sig_sc[b,c] )
// 1 block, 128 threads = 4 wave32; each wave owns 16x16 output tiles.
// A frag (2 VGPR): lane<16 -> K=k0,k0+1 ; lane>=16 -> K=k0+2,k0+3 (M = lane&15)
// B frag (2 VGPR): same K split, N = lane&15
// D (8 VGPR): vgpr r: M=r (lanes 0-15), M=r+8 (lanes 16-31); N = lane&15
// ---------------------------------------------------------------------------
__global__ __launch_bounds__(128) void k_mlp(const float* __restrict__ sq_all,
                                             const float* __restrict__ w1,
                                             const float* __restrict__ b1,
                                             const float* __restrict__ w2,
                                             const float* __restrict__ b2,
                                             const float* __restrict__ sig_sc,
                                             float* __restrict__ sq_w) {
    __shared__ float hbuf[32][32];
    const int tid  = threadIdx.x;
    const int wv   = tid >> 5;
    const int lane = tid & 31;
    const int lr   = lane & 15;
    const int hi   = lane >> 4;

    // ---- GEMM1: 2x2 tiles of 16x16, one per wave; K = 512 ----
    {
        const int mt = wv >> 1, nt = wv & 1;
        const int arow = mt * 16 + lr;      // batch row
        const int ncol = nt * 16 + lr;      // hidden (RED) column
        v8f acc = {};
        for (int k0 = 0; k0 < CC; k0 += 4) {
            const int ka = k0 + hi * 2;
            v2f a, bf;
            a.x  = sq_all[arow * CC + ka];
            a.y  = sq_all[arow * CC + ka + 1];
            bf.x = w1[ncol * CC + ka];       // B[k,n] = w1[n,k]
            bf.y = w1[ncol * CC + ka + 1];
            acc = __builtin_amdgcn_wmma_f32_16x16x4_f32(
                false, a, false, bf, (short)0, acc, false, false);
        }
        const float bias = b1[ncol];
        #pragma unroll
        for (int r = 0; r < 8; ++r) {
            const int m = mt * 16 + r + hi * 8;
            const float v = acc[r] + bias;
            hbuf[m][ncol] = v > 0.0f ? v : 0.0f;   // relu
        }
    }
    __syncthreads();

    // ---- GEMM2: 2x32 tiles of 16x16 over 4 waves; K = 32 ----
    for (int t = wv; t < 64; t += 4) {
        const int mt = t & 1, nt = t >> 1;
        const int arow = mt * 16 + lr;      // batch row
        const int cg   = nt * 16 + lr;      // output channel
        v8f acc = {};
        #pragma unroll
        for (int k0 = 0; k0 < RED; k0 += 4) {
            const int ka = k0 + hi * 2;
            v2f a, bf;
            a.x  = hbuf[arow][ka];
            a.y  = hbuf[arow][ka + 1];
            bf.x = w2[cg * RED + ka];        // B[k,n] = w2[n,k]
            bf.y = w2[cg * RED + ka + 1];
            acc = __builtin_amdgcn_wmma_f32_16x16x4_f32(
                false, a, false, bf, (short)0, acc, false, false);
        }
        const float bias = b2[cg];
        #pragma unroll
        for (int r = 0; r < 8; ++r) {
            const int m = mt * 16 + r + hi * 8;           // batch index
            const float v = acc[r] + bias + sig_sc[m * CC + cg];
            sq_w[m * CC + cg] = v > 0.0f ? v : 0.0f;      // relu
        }
    }
}

// ---------------------------------------------------------------------------
// K4: spatial 7x7 conv (2->1 ch, pad 3) + inference BN + sigmoid(ss1*ss2)
// ---------------------------------------------------------------------------
__global__ __launch_bounds__(256) void k_spatial(const float* __restrict__ ss1,
                                                 const float* __restrict__ ss2,
                                                 const float* __restrict__ cw,
                                                 const float* __restrict__ g,
                                                 const float* __restrict__ bt,
                                                 const float* __restrict__ mn,
                                                 const float* __restrict__ vr,
                                                 float* __restrict__ sp_w) {
    const int idx = blockIdx.x * 256 + threadIdx.x;
    if (idx >= BB * HWP) return;
    const int b  = idx / HWP;
    const int hw = idx - b * HWP;
    const int h  = hw / WW;
    const int w  = hw - h * WW;

    const float* __restrict__ s1 = ss1 + (size_t)b * HWP;
    const float* __restrict__ s2 = ss2 + (size_t)b * HWP;

    float conv = 0.0f;
    #pragma unroll
    for (int kh = 0; kh < 7; ++kh) {
        const int ih = h + kh - 3;
        if (ih < 0 || ih >= HH) continue;
        #pragma unroll
        for (int kw = 0; kw < 7; ++kw) {
            const int iw = w + kw - 3;
            if (iw < 0 || iw >= WW) continue;
            const int p = ih * WW + iw;
            conv += s1[p] * cw[kh * 7 + kw] + s2[p] * cw[49 + kh * 7 + kw];
        }
    }
    const float scale = g[0] * rsqrtf(vr[0] + BN_EPS);
    const float bn = (conv - mn[0]) * scale + bt[0];
    sp_w[idx] = bn + sigmoidf_(s1[hw] * s2[hw]);
}

// ---------------------------------------------------------------------------
// K5: streaming apply: out = x * (1 + sigmoid(sq_w[b,c] * sp_w[b,hw]))
// grid: (C, B); block 256; float4 vectorized. sp_w (400 KB) stays in L2.
// ---------------------------------------------------------------------------
__global__ __launch_bounds__(256) void k_apply(const float4* __restrict__ x,
                                               const float* __restrict__ sq_w,
                                               const float4* __restrict__ sp_w,
                                               float4* __restrict__ out) {
    const int c = blockIdx.x, b = blockIdx.y, tid = threadIdx.x;
    const float qw = sq_w[b * CC + c];
    const size_t plane = (size_t)(b * CC + c) * HW4;
    const float4* __restrict__ xp = x + plane;
    float4*       __restrict__ op = out + plane;
    const float4* __restrict__ sp = sp_w + (size_t)b * HW4;

    for (int i = tid; i < HW4; i += 256) {
        const float4 xv = xp[i];
        const float4 sv = sp[i];
        float4 o;
        o.x = xv.x * (1.0f + sigmoidf_(qw * sv.x));
        o.y = xv.y * (1.0f + sigmoidf_(qw * sv.y));
        o.z = xv.z * (1.0f + sigmoidf_(qw * sv.z));
        o.w = xv.w * (1.0f + sigmoidf_(qw * sv.w));
        op[i] = o;
    }
}

// ---------------------------------------------------------------------------
extern "C" void kernel_launch(void* const* d_in, const int* in_sizes, int n_in,
                              void* d_out, int out_size, void* d_ws, size_t ws_size,
                              hipStream_t stream) {
    const float* x      = (const float*)d_in[0];
    const float* w1     = (const float*)d_in[1];
    const float* b1     = (const float*)d_in[2];
    const float* w2     = (const float*)d_in[3];
    const float* b2     = (const float*)d_in[4];
    const float* conv_w = (const float*)d_in[5];
    const float* bn_g   = (const float*)d_in[6];
    const float* bn_b   = (const float*)d_in[7];
    const float* bn_m   = (const float*)d_in[8];
    const float* bn_v   = (const float*)d_in[9];
    float* out = (float*)d_out;

    // workspace carve-up (floats)
    float* ws      = (float*)d_ws;
    float* ch_sum  = ws;                       // B*C
    float* ch_max  = ch_sum + BB * CC;         // B*C
    float* ss1     = ch_max + BB * CC;         // B*HW
    float* ss2     = ss1    + BB * HWP;        // B*HW
    float* sq_all  = ss2    + BB * HWP;        // B*C
    float* sig_sc  = sq_all + BB * CC;         // B*C
    float* sq_w    = sig_sc + BB * CC;         // B*C
    float* sp_w    = sq_w   + BB * CC;         // B*HW

    k_init<<<(BB * CC + 255) / 256, 256, 0, stream>>>(ch_sum, ch_max);

    dim3 gs((HWP + 255) / 256, BB);
    k_stats<<<gs, 256, 0, stream>>>(x, ch_sum, ch_max, ss1, ss2);

    k_sort<<<BB, 256, 0, stream>>>(ch_sum, ch_max, sig_sc, sq_all);

    k_mlp<<<1, 128, 0, stream>>>(sq_all, w1, b1, w2, b2, sig_sc, sq_w);

    k_spatial<<<(BB * HWP + 255) / 256, 256, 0, stream>>>(
        ss1, ss2, conv_w, bn_g, bn_b, bn_m, bn_v, sp_w);

    dim3 ga(CC, BB);
    k_apply<<<ga, 256, 0, stream>>>((const float4*)x, sq_w,
                                    (const float4*)sp_w, (float4*)out);
}